// Photodetector_36532991819911
// MI455X (gfx1250) — compile-verified
//
#include <hip/hip_runtime.h>

typedef __attribute__((ext_vector_type(2))) float v2f;
typedef __attribute__((ext_vector_type(8))) float v8f;

namespace {
constexpr int TLEN    = 16384;
constexpr int BCOLS   = 1024;
constexpr int JLEN    = 16388;          // TLEN + K - 1
constexpr int CHUNKS  = 64;
constexpr int CHUNK_L = 256;
constexpr int TSTART  = 5;

constexpr float SHOT2 = 1.602176563e-8f;   // 2 * Q_E * BITRATE
constexpr float THERM = 8.2838928e-16f;    // 4 * K_B * T_K * BITRATE / LOAD_R
constexpr float RESP  = 1.0f;
constexpr float DARK  = 1e-10f;
}

// ---------------------------------------------------------------------------
// Kernel 1: per-time batch mean of signal -> noise_sd[0..16387]
// ---------------------------------------------------------------------------
__global__ void k_mean_nsd(const float* __restrict__ sig, float* __restrict__ nsd) {
  __shared__ float red[256];
  const int row = blockIdx.x;          // 0..16383  -> j = row + 4
  const int tid = threadIdx.x;
  const float* rp = sig + (size_t)row * BCOLS;
  float s = rp[tid] + rp[tid + 256] + rp[tid + 512] + rp[tid + 768];
  red[tid] = s;
  __syncthreads();
  for (int off = 128; off > 0; off >>= 1) {
    if (tid < off) red[tid] += red[tid + off];
    __syncthreads();
  }
  if (tid == 0) {
    float mean = RESP * red[0] * (1.0f / 1024.0f);
    nsd[row + 4] = sqrtf(SHOT2 * (mean + DARK) + THERM);
  }
  if (row == 0 && tid < 4) {           // padded rows: mean == 0
    nsd[tid] = sqrtf(SHOT2 * DARK + THERM);
  }
}

// ---------------------------------------------------------------------------
// Kernel 2: FIR.  Each thread owns one batch column b and a 512-step t range,
// streaming its own noise row with a 5-deep sliding window.
// ---------------------------------------------------------------------------
__global__ void k_fir(const float* __restrict__ sig, const float* __restrict__ bco,
                      const float* __restrict__ nsd, const float* __restrict__ noise,
                      float* __restrict__ sout) {
  const int b  = blockIdx.x * 256 + threadIdx.x;   // 0..1023
  const int t0 = blockIdx.y * 512;
  const float b0c = bco[0], b1c = bco[1], b2c = bco[2], b3c = bco[3], b4c = bco[4];
  const float* nrow = noise + (size_t)b * JLEN;

  float x0, x1, x2, x3;
  {
    int j;
    j = t0 + 0; x0 = nsd[j] * nrow[j] + ((j >= 4) ? RESP * sig[(size_t)(j - 4) * BCOLS + b] : 0.0f);
    j = t0 + 1; x1 = nsd[j] * nrow[j] + ((j >= 4) ? RESP * sig[(size_t)(j - 4) * BCOLS + b] : 0.0f);
    j = t0 + 2; x2 = nsd[j] * nrow[j] + ((j >= 4) ? RESP * sig[(size_t)(j - 4) * BCOLS + b] : 0.0f);
    j = t0 + 3; x3 = nsd[j] * nrow[j] + ((j >= 4) ? RESP * sig[(size_t)(j - 4) * BCOLS + b] : 0.0f);
  }
  for (int t = t0; t < t0 + 512; ++t) {
    const int j = t + 4;
    const float x4 = nsd[j] * nrow[j] + RESP * sig[(size_t)t * BCOLS + b];
    const float y = b0c * x0 + b1c * x1 + b2c * x2 + b3c * x3 + b4c * x4;
    sout[(size_t)t * BCOLS + b] = y;
    x0 = x1; x1 = x2; x2 = x3; x3 = x4;
  }
}

// ---------------------------------------------------------------------------
// Block-IIR via WMMA:  Y(16x16) = H(16x16) * S(16x16) + G(16x4) * Z(4x16)
// using 5 chained V_WMMA_F32_16X16X4_F32 per 16-timestep block.
// ---------------------------------------------------------------------------
__device__ __forceinline__ float hsel(const float* h, int i) {
  return (i >= 0 && i < 16) ? h[i] : 0.0f;
}

template <bool PASS_A>
__global__ void k_iir(const float* __restrict__ a_rev, float* __restrict__ sbuf,
                      const float* __restrict__ zInit, float* __restrict__ zA) {
  const int lane  = threadIdx.x & 31;
  const int wv    = threadIdx.x >> 5;
  const int b0    = blockIdx.x * 128 + wv * 16;    // 16-column tile per wave
  const int c     = blockIdx.y;                    // chunk id
  const int mcol  = lane & 15;
  const int khalf = lane >> 4;                     // 0 | 1
  const int kh    = khalf * 2;                     // 0 | 2

  // a_rev = [a4,a3,a2,a1,a0=1]
  const float a1 = a_rev[3], a2 = a_rev[2], a3 = a_rev[1], a4 = a_rev[0];

  // impulse response h[0..15]
  float h[16];
  h[0] = 1.0f;
  h[1] = -a1 * h[0];
  h[2] = -(a1 * h[1] + a2 * h[0]);
  h[3] = -(a1 * h[2] + a2 * h[1] + a3 * h[0]);
#pragma unroll
  for (int d = 4; d < 16; ++d)
    h[d] = -(a1 * h[d - 1] + a2 * h[d - 2] + a3 * h[d - 3] + a4 * h[d - 4]);

  // A-operands (16x4 f32 layout: v0 = A[m][kh], v1 = A[m][kh+1])
  v2f aH0, aH1, aH2, aH3, aG;
  aH0.x = hsel(h, mcol - 0 - kh);  aH0.y = hsel(h, mcol - 0 - kh - 1);
  aH1.x = hsel(h, mcol - 4 - kh);  aH1.y = hsel(h, mcol - 4 - kh - 1);
  aH2.x = hsel(h, mcol - 8 - kh);  aH2.y = hsel(h, mcol - 8 - kh - 1);
  aH3.x = hsel(h, mcol - 12 - kh); aH3.y = hsel(h, mcol - 12 - kh - 1);

  const float av[5] = {0.0f, a1, a2, a3, a4};
  auto Gv = [&](int m, int k0) -> float {   // G[m][k0] = -sum_{i=k0+1..4} a_i h[m-i+k0+1]
    float s = 0.0f;
    for (int i = k0 + 1; i <= 4; ++i) s -= av[i] * hsel(h, m - i + k0 + 1);
    return s;
  };
  aG.x = Gv(mcol, kh);
  aG.y = Gv(mcol, kh + 1);

  // recurrent state as a B-operand (4x16): v0 = z[kh], v1 = z[kh+1]
  v2f zB;
  if (PASS_A) {
    zB.x = 0.0f; zB.y = 0.0f;
  } else {
    const v2f* zi2 = (const v2f*)zInit;
    zB = zi2[((size_t)c * BCOLS + b0 + mcol) * 2 + khalf];
  }

  const int T0 = TSTART + c * CHUNK_L;
  const v8f czero = {0.f, 0.f, 0.f, 0.f, 0.f, 0.f, 0.f, 0.f};
  v8f acc = czero;

  // Main loop: no clamping, no store predication.  Per-lane base pointers are
  // advanced by 16 rows per block-step; all row accesses are immediate offsets.
  // Only the last block of the last chunk (pass C) can touch t >= TLEN; it is
  // handled in the clamped epilogue below.
  const int nbMain = (c == CHUNKS - 1) ? 15 : 16;   // pass A never has c == 63
  const float* sp = sbuf + (size_t)(T0 + kh) * BCOLS + b0 + mcol;
  float*       wp = sbuf + (size_t)(T0 + khalf * 8) * BCOLS + b0 + mcol;

  for (int bs = 0; bs < nbMain; ++bs) {
    v2f sB0, sB1, sB2, sB3;
    sB0.x = sp[0 * BCOLS];  sB0.y = sp[1 * BCOLS];
    sB1.x = sp[4 * BCOLS];  sB1.y = sp[5 * BCOLS];
    sB2.x = sp[8 * BCOLS];  sB2.y = sp[9 * BCOLS];
    sB3.x = sp[12 * BCOLS]; sB3.y = sp[13 * BCOLS];

    // Y = G*Z + H*S  (five chained f32 WMMAs, accumulate in C)
    acc = __builtin_amdgcn_wmma_f32_16x16x4_f32(false, aG,  false, zB,  (short)0, czero, false, false);
    acc = __builtin_amdgcn_wmma_f32_16x16x4_f32(false, aH0, false, sB0, (short)0, acc,   false, false);
    acc = __builtin_amdgcn_wmma_f32_16x16x4_f32(false, aH1, false, sB1, (short)0, acc,   false, false);
    acc = __builtin_amdgcn_wmma_f32_16x16x4_f32(false, aH2, false, sB2, (short)0, acc,   false, false);
    acc = __builtin_amdgcn_wmma_f32_16x16x4_f32(false, aH3, false, sB3, (short)0, acc,   false, false);

    if (!PASS_A) {
      // C/D layout: VGPR r = row (r + 8*khalf), col = lane&15 — all rows valid here
      wp[0 * BCOLS] = acc[0]; wp[1 * BCOLS] = acc[1];
      wp[2 * BCOLS] = acc[2]; wp[3 * BCOLS] = acc[3];
      wp[4 * BCOLS] = acc[4]; wp[5 * BCOLS] = acc[5];
      wp[6 * BCOLS] = acc[6]; wp[7 * BCOLS] = acc[7];
      wp += 16 * BCOLS;
    }

    // next-block state, D-layout -> B-layout (shuffles hoisted: full EXEC)
    const float sh7 = __shfl(acc[7], lane + 16, 32);
    const float sh6 = __shfl(acc[6], lane + 16, 32);
    zB.x = (lane < 16) ? sh7 : acc[5];
    zB.y = (lane < 16) ? sh6 : acc[4];

    sp += 16 * BCOLS;
  }

  if (PASS_A) {
    // final chunk state z[k] = y_{end-k} = D rows 15..12 (VGPRs 7..4, hi lanes)
    if (lane >= 16) {
      float4 st; st.x = acc[7]; st.y = acc[6]; st.z = acc[5]; st.w = acc[4];
      ((float4*)zA)[(size_t)c * BCOLS + b0 + (lane - 16)] = st;
    }
  } else if (c == CHUNKS - 1) {
    // Epilogue: partial block t = 16373..16388 (11 valid rows), clamped reads.
    // H is lower-triangular, so garbage rows only feed outputs we never store.
    const int t0 = T0 + 15 * 16;
    v2f sB0, sB1, sB2, sB3;
    {
      int r;
      r = t0 + 0  + kh; { int ra = r > 16383 ? 16383 : r, rb = r + 1 > 16383 ? 16383 : r + 1;
        sB0.x = sbuf[(size_t)ra * BCOLS + b0 + mcol]; sB0.y = sbuf[(size_t)rb * BCOLS + b0 + mcol]; }
      r = t0 + 4  + kh; { int ra = r > 16383 ? 16383 : r, rb = r + 1 > 16383 ? 16383 : r + 1;
        sB1.x = sbuf[(size_t)ra * BCOLS + b0 + mcol]; sB1.y = sbuf[(size_t)rb * BCOLS + b0 + mcol]; }
      r = t0 + 8  + kh; { int ra = r > 16383 ? 16383 : r, rb = r + 1 > 16383 ? 16383 : r + 1;
        sB2.x = sbuf[(size_t)ra * BCOLS + b0 + mcol]; sB2.y = sbuf[(size_t)rb * BCOLS + b0 + mcol]; }
      r = t0 + 12 + kh; { int ra = r > 16383 ? 16383 : r, rb = r + 1 > 16383 ? 16383 : r + 1;
        sB3.x = sbuf[(size_t)ra * BCOLS + b0 + mcol]; sB3.y = sbuf[(size_t)rb * BCOLS + b0 + mcol]; }
    }
    acc = __builtin_amdgcn_wmma_f32_16x16x4_f32(false, aG,  false, zB,  (short)0, czero, false, false);
    acc = __builtin_amdgcn_wmma_f32_16x16x4_f32(false, aH0, false, sB0, (short)0, acc,   false, false);
    acc = __builtin_amdgcn_wmma_f32_16x16x4_f32(false, aH1, false, sB1, (short)0, acc,   false, false);
    acc = __builtin_amdgcn_wmma_f32_16x16x4_f32(false, aH2, false, sB2, (short)0, acc,   false, false);
    acc = __builtin_amdgcn_wmma_f32_16x16x4_f32(false, aH3, false, sB3, (short)0, acc,   false, false);

    const int tb = t0 + khalf * 8;
#pragma unroll
    for (int r = 0; r < 8; ++r) {
      const int t = tb + r;
      if (t < TLEN) sbuf[(size_t)t * BCOLS + b0 + mcol] = acc[r];
    }
  }
}

// ---------------------------------------------------------------------------
// Kernel 4: tiny sequential scan across the 64 chunks.
// z_init[c+1] = zA[c] + A^256 * z_init[c]   (per batch column, 63 steps)
// ---------------------------------------------------------------------------
__global__ void k_scan(const float* __restrict__ a_rev, const float* __restrict__ sbuf,
                       const float* __restrict__ zA, float* __restrict__ zInit) {
  const int b = blockIdx.x * 256 + threadIdx.x;    // 0..1023
  const float a1 = a_rev[3], a2 = a_rev[2], a3 = a_rev[1], a4 = a_rev[0];

  float M[16] = {-a1, -a2, -a3, -a4,
                 1.f, 0.f, 0.f, 0.f,
                 0.f, 1.f, 0.f, 0.f,
                 0.f, 0.f, 1.f, 0.f};
  for (int it = 0; it < 8; ++it) {                 // A^(2^8) = A^256
    float T[16];
#pragma unroll
    for (int i = 0; i < 4; ++i)
#pragma unroll
      for (int j = 0; j < 4; ++j) {
        float acc = 0.f;
#pragma unroll
        for (int k = 0; k < 4; ++k) acc += M[i * 4 + k] * M[k * 4 + j];
        T[i * 4 + j] = acc;
      }
#pragma unroll
    for (int i = 0; i < 16; ++i) M[i] = T[i];
  }

  // initial state at t = 4: (y4,y3,y2,y1) = (s4,s3,s2,s1)
  float z0 = sbuf[(size_t)4 * BCOLS + b];
  float z1 = sbuf[(size_t)3 * BCOLS + b];
  float z2 = sbuf[(size_t)2 * BCOLS + b];
  float z3 = sbuf[(size_t)1 * BCOLS + b];

  float4* zi = (float4*)zInit;
  const float4* za = (const float4*)zA;
  float4 v; v.x = z0; v.y = z1; v.z = z2; v.w = z3;
  zi[b] = v;
  for (int c = 0; c < CHUNKS - 1; ++c) {
    const float4 t = za[(size_t)c * BCOLS + b];
    const float n0 = t.x + M[0]  * z0 + M[1]  * z1 + M[2]  * z2 + M[3]  * z3;
    const float n1 = t.y + M[4]  * z0 + M[5]  * z1 + M[6]  * z2 + M[7]  * z3;
    const float n2 = t.z + M[8]  * z0 + M[9]  * z1 + M[10] * z2 + M[11] * z3;
    const float n3 = t.w + M[12] * z0 + M[13] * z1 + M[14] * z2 + M[15] * z3;
    z0 = n0; z1 = n1; z2 = n2; z3 = n3;
    float4 w; w.x = z0; w.y = z1; w.z = z2; w.w = z3;
    zi[(size_t)(c + 1) * BCOLS + b] = w;
  }
}

// ---------------------------------------------------------------------------
extern "C" void kernel_launch(void* const* d_in, const int* in_sizes, int n_in,
                              void* d_out, int out_size, void* d_ws, size_t ws_size,
                              hipStream_t stream) {
  (void)in_sizes; (void)n_in; (void)out_size; (void)ws_size;
  const float* sig   = (const float*)d_in[0];   // (16384,1024)
  const float* bco   = (const float*)d_in[1];   // (5)
  const float* arev  = (const float*)d_in[2];   // (5)
  const float* noise = (const float*)d_in[3];   // (1024,16388)
  float* out = (float*)d_out;                   // (16384,1024); also holds s in-place

  char* ws = (char*)d_ws;
  float* nsd   = (float*)ws;                                     // 16388 floats
  float* zA    = (float*)(ws + 16400 * 4);                       // 64*1024 float4
  float* zInit = (float*)(ws + 16400 * 4 + 64 * 1024 * 16);      // 64*1024 float4

  // 1) batch means -> noise_sd
  k_mean_nsd<<<dim3(16384), dim3(256), 0, stream>>>(sig, nsd);
  // 2) FIR -> s, written into d_out
  k_fir<<<dim3(4, 32), dim3(256), 0, stream>>>(sig, bco, nsd, noise, out);
  // 3) Pass A: per-chunk zero-init block-IIR (WMMA), store end states
  k_iir<true><<<dim3(8, CHUNKS - 1), dim3(256), 0, stream>>>(arev, out, nullptr, zA);
  // 4) sequential chunk-state scan (reads s rows 1..4 from d_out, still intact)
  k_scan<<<dim3(4), dim3(256), 0, stream>>>(arev, out, zA, zInit);
  // 5) Pass C: per-chunk corrected block-IIR (WMMA), writes y in place over s;
  //    rows 0..4 remain the FIR head, matching the reference
  k_iir<false><<<dim3(8, CHUNKS), dim3(256), 0, stream>>>(arev, out, zInit, zA);
}